// CombinedLoss_85761906966842
// MI455X (gfx1250) — compile-verified
//
#include <hip/hip_runtime.h>
#include <math.h>

// Problem constants (reference: pred (8,4,256,256) f32, target (8,256,256) int)
#define NN 8
#define NC 4
#define CH 3          // channels 1..3 are the only ones whose EDTs are used
#define H  256
#define W  256
#define HW (H*W)
#define NOSEED_F 10000.0f

typedef float v2f_t __attribute__((ext_vector_type(2)));
typedef float v8f_t __attribute__((ext_vector_type(8)));

// ---------------------------------------------------------------------------
// Deterministic wave32 all-reduce (sum) via V_WMMA_F32_16X16X4_F32.
// A-layout (32b, 16x4): lanes0-15 -> {K0 in a0, K1 in a1}, lanes16-31 -> {K2,K3}.
// With a0 = v, a1 = 0, B = ones: D[m][n] = v[m] + v[m+16].
// D-layout: vgpr j holds row M=j (lanes0-15) / M=j+8 (lanes16-31), so each lane's
// 8 D registers cover 8 distinct rows; summing them gives one half-sum per
// half-wave (identical within the half). A second WMMA adds the two halves.
// EXEC must be all-ones across the wave at the call site.
// ---------------------------------------------------------------------------
__device__ __forceinline__ float wave32_sum_wmma(float v) {
  v2f_t ones; ones[0] = 1.0f; ones[1] = 1.0f;
  v8f_t zeroC = {};
  v2f_t a; a[0] = v; a[1] = 0.0f;
  v8f_t d = __builtin_amdgcn_wmma_f32_16x16x4_f32(
      /*neg_a=*/false, a, /*neg_b=*/false, ones,
      /*c_mod=*/(short)0, zeroC, /*reuse_a=*/false, /*reuse_b=*/false);
  float half = ((d[0] + d[1]) + (d[2] + d[3])) + ((d[4] + d[5]) + (d[6] + d[7]));
  v2f_t a2; a2[0] = half; a2[1] = 0.0f;
  v8f_t d2 = __builtin_amdgcn_wmma_f32_16x16x4_f32(
      false, a2, false, ones, (short)0, zeroC, false, false);
  return d2[0];   // total, replicated to every lane
}

// ---------------------------------------------------------------------------
// Kernel 0: softmax over C=4 per pixel, emit pred_seed = (p_c >= 0.5) bytes
// for channels 1..3.  seed layout: [(n*CH + (c-1))*HW + h*W + w]
// ---------------------------------------------------------------------------
__global__ void k0_seed(const float* __restrict__ pred,
                        unsigned char* __restrict__ seed) {
  int idx = blockIdx.x * blockDim.x + threadIdx.x;
  if (idx >= NN * HW) return;
  int n = idx / HW;
  int hw = idx - n * HW;
  const float* px = pred + (size_t)n * NC * HW + hw;
  float x0 = px[0], x1 = px[HW], x2 = px[2 * HW], x3 = px[3 * HW];
  float m = fmaxf(fmaxf(x0, x1), fmaxf(x2, x3));
  float den = expf(x0 - m) + expf(x1 - m) + expf(x2 - m) + expf(x3 - m);
  float lse = m + logf(den);
  seed[(size_t)(n * CH + 0) * HW + hw] = (expf(x1 - lse) >= 0.5f) ? 1 : 0;
  seed[(size_t)(n * CH + 1) * HW + hw] = (expf(x2 - lse) >= 0.5f) ? 1 : 0;
  seed[(size_t)(n * CH + 2) * HW + hw] = (expf(x3 - lse) >= 0.5f) ? 1 : 0;
}

// ---------------------------------------------------------------------------
// Kernel 1: vertical binary squared DT (reference _sq_dt_binary along H).
// One thread per column; down pass stores dl, up pass recomputes seeds
// (L2-hot re-read) and writes d*d.  blockIdx = fam*(NN*CH) + n*CH + ci,
// fam: 0=fg(tgt==c) 1=bg(tgt!=c) 2=pred-seed.
// ---------------------------------------------------------------------------
__global__ void k1_vert(const int* __restrict__ target,
                        const unsigned char* __restrict__ seedp,
                        float* __restrict__ gfg,
                        float* __restrict__ gbg,
                        float* __restrict__ gpr) {
  int b = blockIdx.x;
  int fam = b / (NN * CH);
  int r = b - fam * (NN * CH);
  int n = r / CH, ci = r - n * CH;
  int c = ci + 1;
  int w = threadIdx.x;

  float* g = (fam == 0) ? gfg : ((fam == 1) ? gbg : gpr);
  float* gcol = g + (size_t)(n * CH + ci) * HW + w;
  const int* tcol = target + (size_t)n * HW + w;
  const unsigned char* scol = seedp + (size_t)(n * CH + ci) * HW + w;

  int last = -1;
  for (int h = 0; h < H; ++h) {
    bool s;
    if (fam == 0)       s = (tcol[h * W] == c);
    else if (fam == 1)  s = (tcol[h * W] != c);
    else                s = (scol[h * W] != 0);
    if (s) last = h;
    gcol[h * W] = (last >= 0) ? (float)(h - last) : NOSEED_F;
  }
  int nxt = -1;
  for (int h = H - 1; h >= 0; --h) {
    bool s;
    if (fam == 0)       s = (tcol[h * W] == c);
    else if (fam == 1)  s = (tcol[h * W] != c);
    else                s = (scol[h * W] != 0);
    if (s) nxt = h;
    float dr = (nxt >= 0) ? (float)(nxt - h) : NOSEED_F;
    float d = fminf(gcol[h * W], dr);
    gcol[h * W] = d * d;
  }
}

// ---------------------------------------------------------------------------
// Kernel 2: horizontal lower envelope (brute force, identical candidate set
// to the reference scan; fp32 min is order-independent) for fg/bg/pred rows
// simultaneously, fused with softmax recompute and CE/BD/HD partial sums.
// One block per (n, c, h) row: blockIdx = (n*CH + ci)*H + h.
// Per-block partial sums reduced deterministically (LDS tree + WMMA wave sum)
// and written to partials[block] (float4: ce, bd, t1, t2).
// ---------------------------------------------------------------------------
__global__ void k2_horiz(const float* __restrict__ pred,
                         const int* __restrict__ target,
                         const float* __restrict__ gfg,
                         const float* __restrict__ gbg,
                         const float* __restrict__ gpr,
                         float4* __restrict__ partials) {
  __shared__ float sfg[W], sbg[W], spr[W];
  __shared__ float4 sred[W];

  int b = blockIdx.x;
  int h = b % H;
  int nc = b / H;
  int n = nc / CH, ci = nc - n * CH;
  int c = ci + 1;
  int j = threadIdx.x;

  size_t rowOff = (size_t)nc * HW + (size_t)h * W;
  sfg[j] = gfg[rowOff + j];
  sbg[j] = gbg[rowOff + j];
  spr[j] = gpr[rowOff + j];
  __syncthreads();

  float dfg = 1e18f, dbg = 1e18f, dpr = 1e18f;
  for (int y = 0; y < W; ++y) {
    float t = (float)(j - y);
    float q = t * t;                       // exact for |j-y| <= 255
    dfg = fminf(dfg, sfg[y] + q);
    dbg = fminf(dbg, sbg[y] + q);
    dpr = fminf(dpr, spr[y] + q);
  }
  float tf = sqrtf(dfg), tb = sqrtf(dbg), tp = sqrtf(dpr);

  const float* px = pred + (size_t)n * NC * HW + (size_t)h * W + j;
  float x0 = px[0], x1 = px[HW], x2 = px[2 * HW], x3 = px[3 * HW];
  float m = fmaxf(fmaxf(x0, x1), fmaxf(x2, x3));
  float den = expf(x0 - m) + expf(x1 - m) + expf(x2 - m) + expf(x3 - m);
  float lse = m + logf(den);
  int tg = target[(size_t)n * HW + (size_t)h * W + j];

  float xc = (c == 1) ? x1 : ((c == 2) ? x2 : x3);
  float pc = expf(xc - lse);

  float ce = 0.0f;
  if (c == 1) {   // count CE once per pixel (this kernel visits pixels 3x)
    float xt = (tg == 0) ? x0 : ((tg == 1) ? x1 : ((tg == 2) ? x2 : x3));
    ce = lse - xt;                          // = -log_p[target]
  }
  float bd = pc * (tf - tb);                // p * signed_dm
  float t1 = pc * (tf * tf);                // p * dist_fg^2
  float t2 = (tg == c) ? (tp * tp) : 0.0f;  // gt * dist_pred^2

  sred[j] = make_float4(ce, bd, t1, t2);
  __syncthreads();
  for (int s = 128; s >= 32; s >>= 1) {
    if (j < s) {
      float4 a = sred[j], q2 = sred[j + s];
      sred[j] = make_float4(a.x + q2.x, a.y + q2.y, a.z + q2.z, a.w + q2.w);
    }
    __syncthreads();
  }
  if (j < 32) {   // wave 0 fully active -> EXEC all-ones for WMMA
    float4 v = sred[j];
    float r0 = wave32_sum_wmma(v.x);
    float r1 = wave32_sum_wmma(v.y);
    float r2 = wave32_sum_wmma(v.z);
    float r3 = wave32_sum_wmma(v.w);
    if (j == 0) partials[b] = make_float4(r0, r1, r2, r3);
  }
}

// ---------------------------------------------------------------------------
// Kernel 3: deterministic final reduction (fixed-order strided sums, LDS tree,
// WMMA wave sum) and scalar combine.  out = (total, ce, bd, hd).
// ---------------------------------------------------------------------------
__global__ void k3_final(const float4* __restrict__ partials, int nPart,
                         float* __restrict__ out) {
  __shared__ float4 sred[256];
  int t = threadIdx.x;
  float4 acc = make_float4(0.f, 0.f, 0.f, 0.f);
  for (int i = t; i < nPart; i += 256) {
    float4 v = partials[i];
    acc.x += v.x; acc.y += v.y; acc.z += v.z; acc.w += v.w;
  }
  sred[t] = acc;
  __syncthreads();
  for (int s = 128; s >= 32; s >>= 1) {
    if (t < s) {
      float4 a = sred[t], q = sred[t + s];
      sred[t] = make_float4(a.x + q.x, a.y + q.y, a.z + q.z, a.w + q.w);
    }
    __syncthreads();
  }
  if (t < 32) {
    float4 v = sred[t];
    float Sce = wave32_sum_wmma(v.x);
    float Sbd = wave32_sum_wmma(v.y);
    float St1 = wave32_sum_wmma(v.z);
    float St2 = wave32_sum_wmma(v.w);
    if (t == 0) {
      float ce = Sce / (float)(NN * HW);            // mean over N*H*W
      float bd = Sbd / (float)(NN * CH);            // mean over N*(C-1) row-sums
      float hd = (St1 + St2) / ((float)HW * (2.0f * NN * CH));
      float total = 1.0f * ce + 0.5f * bd + 0.5f * hd;
      out[0] = total; out[1] = ce; out[2] = bd; out[3] = hd;
    }
  }
}

// ---------------------------------------------------------------------------
extern "C" void kernel_launch(void* const* d_in, const int* in_sizes, int n_in,
                              void* d_out, int out_size, void* d_ws, size_t ws_size,
                              hipStream_t stream) {
  (void)in_sizes; (void)n_in; (void)out_size; (void)ws_size;
  const float* pred   = (const float*)d_in[0];
  const int*   target = (const int*)d_in[1];

  char* ws = (char*)d_ws;
  const size_t S = (size_t)NN * CH * HW * sizeof(float);      // 6 MB per family
  float* gfg = (float*)(ws);
  float* gbg = (float*)(ws + S);
  float* gpr = (float*)(ws + 2 * S);
  unsigned char* seed = (unsigned char*)(ws + 3 * S);          // 1.5 MB
  float4* partials = (float4*)(ws + 3 * S + (size_t)NN * CH * HW); // 16B aligned

  k0_seed<<<(NN * HW + 255) / 256, 256, 0, stream>>>(pred, seed);
  k1_vert<<<3 * NN * CH, W, 0, stream>>>(target, seed, gfg, gbg, gpr);
  const int nBlocks2 = NN * CH * H;   // 6144
  k2_horiz<<<nBlocks2, W, 0, stream>>>(pred, target, gfg, gbg, gpr, partials);
  k3_final<<<1, 256, 0, stream>>>(partials, nBlocks2, (float*)d_out);
}